// AllocatorNN_73375221284989
// MI455X (gfx1250) — compile-verified
//
#include <hip/hip_runtime.h>
#include <math.h>

typedef float v2f __attribute__((ext_vector_type(2)));
typedef float v8f __attribute__((ext_vector_type(8)));

#define H 256
#define NUM_BS 128
#define NUE 4096

// LDS A-panel row strides (floats): stride % 64 == 4 -> conflict-free lane
// reads (banks 4*lm), stride*4 % 16 == 0 -> 16B alignment for B128 async copy.
#define LDSA_STRIDE_P 260   // proj:   K=256 panel
#define LDSA_STRIDE_D 516   // decode: K=512 panel

// ---------------------------------------------------------------------------
// Encoder: out = LayerNorm(ReLU(x @ W + b)) ; x: [rows,8], W: [8,256]
// one block (256 threads) per row; block-wide LDS reduction for mean/var
// ---------------------------------------------------------------------------
__global__ void encode_ln_kernel(const float* __restrict__ x,
                                 const float* __restrict__ W,
                                 const float* __restrict__ b,
                                 const float* __restrict__ g,
                                 const float* __restrict__ be,
                                 float* __restrict__ out) {
  __shared__ float s1[H];
  __shared__ float s2[H];
  const int row = blockIdx.x;
  const int c = threadIdx.x;
  float z = b[c];
#pragma unroll
  for (int d = 0; d < 8; ++d) z += x[row * 8 + d] * W[d * H + c];
  z = fmaxf(z, 0.0f);
  s1[c] = z;
  s2[c] = z * z;
  __syncthreads();
  for (int s = 128; s > 0; s >>= 1) {
    if (c < s) { s1[c] += s1[c + s]; s2[c] += s2[c + s]; }
    __syncthreads();
  }
  const float mean = s1[0] * (1.0f / 256.0f);
  const float var  = s2[0] * (1.0f / 256.0f) - mean * mean;
  const float inv  = rsqrtf(var + 1e-5f);
  out[row * H + c] = (z - mean) * inv * g[c] + be[c];
}

// ---------------------------------------------------------------------------
// pbT[k][j] = b_a1[k] + sum_c bs_f[j][c] * w_a_bs[c][k]   (transposed store
// so the attention kernel reads pbT[k*128+j] fully coalesced over j)
// ---------------------------------------------------------------------------
__global__ void proj_bs_T_kernel(const float* __restrict__ bs_f,
                                 const float* __restrict__ w_a_bs,
                                 const float* __restrict__ b_a1,
                                 float* __restrict__ pbT) {
  __shared__ float srow[H];
  const int j = blockIdx.x;
  const int k = threadIdx.x;
  srow[k] = bs_f[j * H + k];
  __syncthreads();
  float acc = b_a1[k];
  for (int c = 0; c < H; ++c) acc += srow[c] * w_a_bs[c * H + k];
  pbT[k * NUM_BS + j] = acc;
}

// ---------------------------------------------------------------------------
// pu = ue_f @ w_a_ue : [4096,256] x [256,256], exact fp32 WMMA.
// The 16-row A panel is shared by the block's 4 waves -> stage it once into
// LDS with per-lane GLOBAL_LOAD_ASYNC_TO_LDS_B128 (ASYNCcnt path), then each
// wave runs 64 V_WMMA_F32_16X16X4_F32 steps reading A from LDS, B from L2.
// ---------------------------------------------------------------------------
__global__ void wmma_proj_ue_kernel(const float* __restrict__ A,
                                    const float* __restrict__ B,
                                    float* __restrict__ C) {
  __shared__ float sA[16 * LDSA_STRIDE_P];
  const int t = threadIdx.x;                  // 128 threads = 4 wave32
  const int lane = t & 31;
  const int wave = t >> 5;
  const int tile = blockIdx.x * 4 + wave;     // 4096 tiles, N-major in M group
  const int m0 = (tile >> 4) * 16;            // same m0 for all 4 waves
  const int n0 = (tile & 15) * 16;
  // cooperative async copy: 16 rows x 256 floats = 1024 B128 chunks
#pragma unroll
  for (int it = 0; it < 8; ++it) {
    const int cid = t + 128 * it;
    const int row = cid >> 6;                 // 64 chunks per row
    const int kc = (cid & 63) * 4;
    const float* gp = A + (m0 + row) * H + kc;
    const unsigned ldsoff = (unsigned)(size_t)(&sA[row * LDSA_STRIDE_P + kc]);
    asm volatile("global_load_async_to_lds_b128 %0, %1, off"
                 :: "v"(ldsoff), "v"(gp) : "memory");
  }
  asm volatile("s_wait_asynccnt 0x0" ::: "memory");
  __syncthreads();

  const int half = lane >> 4;
  const int lm = lane & 15;
  const float* ap = &sA[lm * LDSA_STRIDE_P];  // A row m per lane (LDS)
  const float* Bcol = B + (n0 + lm);          // B col n per lane (global/L2)
  v8f c = {};
  for (int k0 = 0; k0 < H; k0 += 4) {
    const int ka = k0 + 2 * half;             // ISA 16x4 f32 A layout
    v2f a; a.x = ap[ka];            a.y = ap[ka + 1];
    v2f b; b.x = Bcol[ka * H];      b.y = Bcol[(ka + 1) * H];
    c = __builtin_amdgcn_wmma_f32_16x16x4_f32(false, a, false, b,
                                              (short)0, c, false, false);
  }
  float* Crow = C + n0 + lm;
#pragma unroll
  for (int v = 0; v < 8; ++v)
    Crow[(m0 + v + 8 * half) * H] = c[v];     // D row = v + 8*half
}

// ---------------------------------------------------------------------------
// Fused attention score + mask + argmax (softmax elided: argmax-invariant).
// score(i,j) = b_a2 + sum_k relu(pu[i,k] + pbT[k,j] + snr[i,j]*w_snr[k])*w_a2[k]
// Block: 256 threads = 2 ue rows x 128 bs. h tensor never materialized.
// ---------------------------------------------------------------------------
__global__ void attn_argmax_kernel(const float* __restrict__ pu,
                                   const float* __restrict__ pbT,
                                   const float* __restrict__ snr,
                                   const float* __restrict__ dist,
                                   const float* __restrict__ bs_state,
                                   const float* __restrict__ w_a_snr,
                                   const float* __restrict__ w_a2,
                                   const float* __restrict__ b_a2,
                                   int* __restrict__ idx_out,
                                   float* __restrict__ bsnr_out) {
  __shared__ float spu[2 * H];
  __shared__ float sws[H];
  __shared__ float swa[H];
  __shared__ float ssc[256];
  __shared__ int   sj[256];
  const int t = threadIdx.x;
  const int li = t >> 7;                      // 0/1: which ue row in block
  const int j = t & 127;                      // bs index
  const int i = blockIdx.x * 2 + li;
  sws[t] = w_a_snr[t];
  swa[t] = w_a2[t];
  spu[t]     = pu[blockIdx.x * 2 * H + t];
  spu[t + H] = pu[blockIdx.x * 2 * H + H + t];
  __syncthreads();
  const float sij = snr[i * NUM_BS + j];
  const float* prow = spu + li * H;
  float acc = 0.0f;
  for (int k = 0; k < H; ++k)
    acc += fmaxf(prow[k] + pbT[k * NUM_BS + j] + sij * sws[k], 0.0f) * swa[k];
  acc += b_a2[0];
  if (dist[i * NUM_BS + j] > bs_state[j * 8 + 4]) acc += -1e9f;
  ssc[t] = acc;
  sj[t] = j;
  __syncthreads();
  for (int s = 64; s > 0; s >>= 1) {          // per-half argmax, first-max ties
    if (j < s) {
      const float so = ssc[t + s]; const int jo = sj[t + s];
      if (so > ssc[t] || (so == ssc[t] && jo < sj[t])) { ssc[t] = so; sj[t] = jo; }
    }
    __syncthreads();
  }
  if (j == 0) {
    const int best = sj[t];
    idx_out[i] = best;
    bsnr_out[i] = snr[i * NUM_BS + best];
  }
}

// ---------------------------------------------------------------------------
// x1 = relu(cat([ue_f, bs_f[idx], bsnr]) @ w_d1 + b_d1) : K = 513.
// +1 column handled as rank-1 accumulator init; bs_f[idx] gather folded into
// the async A-panel copy. A panel (16x512) staged once per block via
// GLOBAL_LOAD_ASYNC_TO_LDS_B128; 128 WMMA K-steps per wave.
// ---------------------------------------------------------------------------
__global__ void wmma_decode1_kernel(const float* __restrict__ ue_f,
                                    const float* __restrict__ bs_f,
                                    const int* __restrict__ idx,
                                    const float* __restrict__ bsnr,
                                    const float* __restrict__ w_d1,
                                    const float* __restrict__ b_d1,
                                    float* __restrict__ x1) {
  __shared__ float sA[16 * LDSA_STRIDE_D];
  const int t = threadIdx.x;
  const int lane = t & 31;
  const int wave = t >> 5;
  const int tile = blockIdx.x * 4 + wave;
  const int m0 = (tile >> 4) * 16;
  const int n0 = (tile & 15) * 16;
  // cooperative async copy of cat([ue_f, bs_f[idx]]) panel: 2048 B128 chunks
#pragma unroll
  for (int it = 0; it < 16; ++it) {
    const int cid = t + 128 * it;
    const int row = cid >> 7;                 // 128 chunks per row (512/4)
    const int kc = (cid & 127) * 4;
    const float* gp = (kc < H) ? (ue_f + (m0 + row) * H + kc)
                               : (bs_f + idx[m0 + row] * H + (kc - H));
    const unsigned ldsoff = (unsigned)(size_t)(&sA[row * LDSA_STRIDE_D + kc]);
    asm volatile("global_load_async_to_lds_b128 %0, %1, off"
                 :: "v"(ldsoff), "v"(gp) : "memory");
  }
  asm volatile("s_wait_asynccnt 0x0" ::: "memory");
  __syncthreads();

  const int half = lane >> 4;
  const int lm = lane & 15;
  const float* ap = &sA[lm * LDSA_STRIDE_D];
  const float* Bc = w_d1 + (n0 + lm);
  const float wl = w_d1[512 * H + n0 + lm];   // last (snr) row of w_d1
  v8f c;
#pragma unroll
  for (int v = 0; v < 8; ++v) c[v] = bsnr[m0 + v + 8 * half] * wl;
  for (int k0 = 0; k0 < 512; k0 += 4) {
    const int ka = k0 + 2 * half;
    v2f a; a.x = ap[ka];          a.y = ap[ka + 1];
    v2f b; b.x = Bc[ka * H];      b.y = Bc[(ka + 1) * H];
    c = __builtin_amdgcn_wmma_f32_16x16x4_f32(false, a, false, b,
                                              (short)0, c, false, false);
  }
  const float bias = b_d1[n0 + lm];
#pragma unroll
  for (int v = 0; v < 8; ++v)
    x1[(m0 + v + 8 * half) * H + n0 + lm] = fmaxf(c[v] + bias, 0.0f);
}

// ---------------------------------------------------------------------------
// Fused tail: x2 = relu(x1@w_d2+b_d2); logits=x2@w_cls+b_cls; argmax-1;
// bw = min(sigmoid(relu(x2@w_r1+b_r1)@w_r2+b_r2) * ue5, ue5).
// ---------------------------------------------------------------------------
__global__ void heads_kernel(const float* __restrict__ x1,
                             const float* __restrict__ w_d2,
                             const float* __restrict__ b_d2,
                             const float* __restrict__ w_cls,
                             const float* __restrict__ b_cls,
                             const float* __restrict__ w_r1,
                             const float* __restrict__ b_r1,
                             const float* __restrict__ w_r2,
                             const float* __restrict__ b_r2,
                             const float* __restrict__ ue_state,
                             float* __restrict__ out) {
  __shared__ float sx1[H];
  __shared__ float sx2[128];
  __shared__ float slog[129];
  __shared__ float sr1[64];
  const int i = blockIdx.x;
  const int t = threadIdx.x;                  // 128 threads
  sx1[t] = x1[i * H + t];
  sx1[t + 128] = x1[i * H + 128 + t];
  __syncthreads();
  float acc = b_d2[t];
  for (int k = 0; k < H; ++k) acc += sx1[k] * w_d2[k * 128 + t];
  sx2[t] = fmaxf(acc, 0.0f);
  __syncthreads();
  float lg = b_cls[t];
  for (int k = 0; k < 128; ++k) lg += sx2[k] * w_cls[k * 129 + t];
  slog[t] = lg;
  if (t == 0) {
    float l128 = b_cls[128];
    for (int k = 0; k < 128; ++k) l128 += sx2[k] * w_cls[k * 129 + 128];
    slog[128] = l128;
  }
  if (t < 64) {
    float r = b_r1[t];
    for (int k = 0; k < 128; ++k) r += sx2[k] * w_r1[k * 64 + t];
    sr1[t] = fmaxf(r, 0.0f);
  }
  __syncthreads();
  if (t == 0) {
    int arg = 0;
    float best = slog[0];
    for (int k = 1; k < 129; ++k)
      if (slog[k] > best) { best = slog[k]; arg = k; }   // first-max rule
    out[i] = (float)(arg - 1);
    float r = b_r2[0];
    for (int k = 0; k < 64; ++k) r += sr1[k] * w_r2[k];
    const float sig = 1.0f / (1.0f + expf(-r));
    const float ue5 = ue_state[i * 8 + 4];
    out[NUE + i] = fminf(sig * ue5, ue5);
  }
}

// ---------------------------------------------------------------------------
extern "C" void kernel_launch(void* const* d_in, const int* in_sizes, int n_in,
                              void* d_out, int out_size, void* d_ws, size_t ws_size,
                              hipStream_t stream) {
  const float* ue_state  = (const float*)d_in[0];
  const float* bs_state  = (const float*)d_in[1];
  const float* distances = (const float*)d_in[2];
  const float* snr_db    = (const float*)d_in[3];
  const float* w_ue1 = (const float*)d_in[4];
  const float* b_ue1 = (const float*)d_in[5];
  const float* g_ue  = (const float*)d_in[6];
  const float* be_ue = (const float*)d_in[7];
  const float* w_bs1 = (const float*)d_in[8];
  const float* b_bs1 = (const float*)d_in[9];
  const float* g_bs  = (const float*)d_in[10];
  const float* be_bs = (const float*)d_in[11];
  const float* w_a_ue  = (const float*)d_in[12];
  const float* w_a_bs  = (const float*)d_in[13];
  const float* w_a_snr = (const float*)d_in[14];
  const float* b_a1    = (const float*)d_in[15];
  const float* w_a2    = (const float*)d_in[16];
  const float* b_a2    = (const float*)d_in[17];
  const float* w_d1 = (const float*)d_in[18];
  const float* b_d1 = (const float*)d_in[19];
  const float* w_d2 = (const float*)d_in[20];
  const float* b_d2 = (const float*)d_in[21];
  const float* w_cls = (const float*)d_in[22];
  const float* b_cls = (const float*)d_in[23];
  const float* w_r1  = (const float*)d_in[24];
  const float* b_r1  = (const float*)d_in[25];
  const float* w_r2  = (const float*)d_in[26];
  const float* b_r2  = (const float*)d_in[27];

  // workspace carve (~12.9 MB)
  float* ue_f = (float*)d_ws;                 // [4096,256]
  float* pu   = ue_f + NUE * H;               // [4096,256]
  float* x1   = pu   + NUE * H;               // [4096,256]
  float* bs_f = x1   + NUE * H;               // [128,256]
  float* pbT  = bs_f + NUM_BS * H;            // [256,128]
  float* bsnr = pbT  + H * NUM_BS;            // [4096]
  int*   idx  = (int*)(bsnr + NUE);           // [4096]

  encode_ln_kernel<<<NUE, H, 0, stream>>>(ue_state, w_ue1, b_ue1, g_ue, be_ue, ue_f);
  encode_ln_kernel<<<NUM_BS, H, 0, stream>>>(bs_state, w_bs1, b_bs1, g_bs, be_bs, bs_f);
  proj_bs_T_kernel<<<NUM_BS, H, 0, stream>>>(bs_f, w_a_bs, b_a1, pbT);
  wmma_proj_ue_kernel<<<(NUE / 16) * (H / 16) / 4, 128, 0, stream>>>(ue_f, w_a_ue, pu);
  attn_argmax_kernel<<<NUE / 2, 256, 0, stream>>>(pu, pbT, snr_db, distances, bs_state,
                                                  w_a_snr, w_a2, b_a2, idx, bsnr);
  wmma_decode1_kernel<<<(NUE / 16) * (H / 16) / 4, 128, 0, stream>>>(ue_f, bs_f, idx, bsnr,
                                                                     w_d1, b_d1, x1);
  heads_kernel<<<NUE, 128, 0, stream>>>(x1, w_d2, b_d2, w_cls, b_cls,
                                        w_r1, b_r1, w_r2, b_r2, ue_state, (float*)d_out);
}